// PamDelayModel_22393959481701
// MI455X (gfx1250) — compile-verified
//
#include <hip/hip_runtime.h>

#define DT_F 0.01f

// 8-point LUT linear interpolation matching jnp searchsorted(side='left') semantics.
// xp/fp point into LDS -> dynamic indexing lowers to ds_load_b32 (no scratch spill).
__device__ __forceinline__ float interp8(float x, const float* __restrict__ xp,
                                         const float* __restrict__ fp) {
    float xc = fminf(fmaxf(x, xp[0]), xp[7]);
    int idx = 0;
#pragma unroll
    for (int i = 0; i < 8; ++i) idx += (xp[i] < xc) ? 1 : 0;   // count strictly-less == searchsorted left
    idx = idx < 1 ? 1 : (idx > 7 ? 7 : idx);                   // clip to [1, 7]
    float x0 = xp[idx - 1], x1 = xp[idx];
    float f0 = fp[idx - 1], f1 = fp[idx];
    float t = (xc - x0) / (x1 - x0 + 1e-12f);
    return f0 + t * (f1 - f0);
}

// Fractional ring-buffer read index for one element; returns alpha, clamped idx0, idx1.
__device__ __forceinline__ void ring_index(float L, int wp, int buf_len,
                                           int& idx0c, int& idx1, float& alpha) {
    float D = L / DT_F;                                        // L / DT, f32 division as in ref
    D = fminf(fmaxf(D, 0.0f), (float)(buf_len - 2));
    float r = (float)wp - D;
    if (r < 0.0f) r += (float)buf_len;                         // == jnp.mod for |r| < buf_len
    int idx0 = (int)floorf(r);
    alpha = r - (float)idx0;
    idx1 = idx0 + 1;
    if (idx1 >= buf_len) idx1 -= buf_len;                      // (idx0+1) % buf_len (idx0 may be buf_len)
    idx0c = idx0 > (buf_len - 1) ? (buf_len - 1) : idx0;       // XLA gather clamp
}

__global__ __launch_bounds__(256) void pam_delay_kernel(
    const float* __restrict__ target,     // [N*C]
    const float* __restrict__ buffer,     // [N*C, buf_len] ring buffer (read-only)
    const float* __restrict__ current,    // [N*C]
    const float* __restrict__ tau_x, const float* __restrict__ tau_f,    // [8],[8]
    const float* __restrict__ dead_x, const float* __restrict__ dead_f,  // [8],[8]
    const int* __restrict__ wp_ptr,       // [1]
    float* __restrict__ out,              // [N*C]
    int total, int buf_len)
{
    // Stage the four 8-entry LUTs (128 B) into LDS once per block.
    __shared__ float lut[32];
    const int t = threadIdx.x;
    if (t < 32) {
        const float* src = (t < 8) ? tau_x : (t < 16) ? tau_f : (t < 24) ? dead_x : dead_f;
        lut[t] = src[t & 7];
    }
    __syncthreads();

    // Two elements per thread: e0 = 2*pair, e1 = e0+1 -> b64 coalesced streaming
    // accesses and 4 independent gathers in flight per thread.
    const int pair = blockIdx.x * 256 + t;
    const int e0 = pair * 2;
    if (e0 >= total) return;
    const bool has2 = (e0 + 1) < total;

    const int wp = *wp_ptr;                       // wave-uniform scalar load

    float xa, xb;
    if (has2) {
        const float2 tp = *(const float2*)(target + e0);
        xa = tp.x; xb = tp.y;
    } else {
        xa = target[e0]; xb = xa;
    }

    const float* row0 = buffer + (size_t)e0 * (size_t)buf_len;
    const float* row1 = row0 + buf_len;

    // Warm both lanes' ring rows while the LUT math resolves the gather offsets
    // (lowers to global_prefetch_b8 on gfx1250).
    __builtin_prefetch(row0 + 48, 0, 0);
    __builtin_prefetch(row1 + 48, 0, 0);

    // Dead-time LUT -> fractional read indices for both elements.
    const float La = interp8(xa, lut + 16, lut + 24);
    const float Lb = interp8(xb, lut + 16, lut + 24);
    int i0a, i1a, i0b, i1b; float alA, alB;
    ring_index(La, wp, buf_len, i0a, i1a, alA);
    ring_index(Lb, wp, buf_len, i0b, i1b, alB);

    // Issue all four gathers back-to-back (the only heavy memory op).
    float v0a = row0[i0a];
    float v1a = row0[i1a];
    float v0b = row1[i0b];
    float v1b = row1[i1b];
    if (i0a == wp) v0a = xa;                      // slot wp holds the fresh write of target
    if (i1a == wp) v1a = xa;
    if (i0b == wp) v0b = xb;
    if (i1b == wp) v1b = xb;

    // Overlap the second LUT + current load with the gathers in flight.
    float ca, cb;
    if (has2) {
        const float2 cp = *(const float2*)(current + e0);
        ca = cp.x; cb = cp.y;
    } else {
        ca = current[e0]; cb = ca;
    }
    const float tauA = interp8(xa, lut + 0, lut + 8);
    const float tauB = interp8(xb, lut + 0, lut + 8);

    const float delA = (1.0f - alA) * v0a + alA * v1a;
    const float delB = (1.0f - alB) * v0b + alB * v1b;
    const float afA = DT_F / (tauA + DT_F);
    const float afB = DT_F / (tauB + DT_F);
    const float oA = (1.0f - afA) * ca + afA * delA;
    const float oB = (1.0f - afB) * cb + afB * delB;

    if (has2) {
        *(float2*)(out + e0) = make_float2(oA, oB);
    } else {
        out[e0] = oA;
    }
}

extern "C" void kernel_launch(void* const* d_in, const int* in_sizes, int n_in,
                              void* d_out, int out_size, void* d_ws, size_t ws_size,
                              hipStream_t stream) {
    const float* target  = (const float*)d_in[0];
    const float* buffer  = (const float*)d_in[1];
    const float* current = (const float*)d_in[2];
    const float* tau_x   = (const float*)d_in[3];
    const float* tau_f   = (const float*)d_in[4];
    const float* dead_x  = (const float*)d_in[5];
    const float* dead_f  = (const float*)d_in[6];
    const int*   wp      = (const int*)d_in[7];
    float* out = (float*)d_out;

    const int total   = in_sizes[0];                 // N*C = 262144
    const int buf_len = in_sizes[1] / in_sizes[0];   // 102

    const int pairs  = (total + 1) / 2;
    const int blocks = (pairs + 255) / 256;
    pam_delay_kernel<<<blocks, 256, 0, stream>>>(target, buffer, current,
                                                 tau_x, tau_f, dead_x, dead_f,
                                                 wp, out, total, buf_len);
}